// EdgeAlignmentModule_6528350290169
// MI455X (gfx1250) — compile-verified
//
#include <hip/hip_runtime.h>
#include <stdint.h>

// ---------------------------------------------------------------------------
// EdgeAlignmentModule on MI455X (gfx1250)
//
// Replaces the reference's 400MB dense node-pair table with an 8MB
// open-addressing hash table in d_ws. Latency-bound random probes are
// overlapped with CDNA5 async global->LDS staging of the streaming
// edge_attr_new tile; output rows are written with b128 stores.
// ---------------------------------------------------------------------------

#define TPB 256

__device__ __forceinline__ unsigned hash_mix(int k) {
    unsigned x = (unsigned)k;
    x ^= x >> 16;
    x *= 0x7feb352dU;
    x ^= x >> 15;
    x *= 0x846ca68bU;
    x ^= x >> 16;
    return x;
}

// ---- Kernel 1: reset hash table (keys and vals to -1) ----------------------
__global__ void hash_init_kernel(int4* __restrict__ p, int n4) {
    int i = blockIdx.x * blockDim.x + threadIdx.x;
    if (i < n4) p[i] = make_int4(-1, -1, -1, -1);
}

// ---- Kernel 2: insert old edges into hash table ----------------------------
// Deterministic under races: CAS claims a slot for a key; duplicate keys
// resolve via atomicMax on the value (max old-edge index wins every run).
__global__ void hash_insert_kernel(const int* __restrict__ ei_old,
                                   const int* __restrict__ nptr,
                                   int E_old,
                                   int* __restrict__ hkey,
                                   int* __restrict__ hval,
                                   unsigned hmask) {
    int e = blockIdx.x * blockDim.x + threadIdx.x;
    if (e >= E_old) return;
    const int N   = *nptr;
    const int src = ei_old[e];
    const int dst = ei_old[E_old + e];
    const int key = src * N + dst;   // matches reference int32 key

    unsigned s = hash_mix(key) & hmask;
    for (unsigned probes = 0; probes <= hmask; ++probes) {
        int prev = atomicCAS(&hkey[s], -1, key);
        if (prev == -1 || prev == key) {
            atomicMax(&hval[s], e);
            return;
        }
        s = (s + 1) & hmask;
    }
    // table full (cannot happen when hsize > E_old): drop edge, never hang
}

// ---- Kernel 3: lookup + gather + emit --------------------------------------
__global__ void edge_align_kernel(const int*   __restrict__ ei_new,
                                  const float* __restrict__ attr_new,
                                  const float* __restrict__ attr_old,
                                  const float* __restrict__ flow_old,
                                  const int*   __restrict__ hkey,
                                  const int*   __restrict__ hval,
                                  const int*   __restrict__ nptr,
                                  int E_new,
                                  unsigned hmask,
                                  float* __restrict__ out) {
    __shared__ float tile[TPB * 3];   // 3 KB staging of edge_attr_new

    const int t          = threadIdx.x;
    const int blockStart = blockIdx.x * TPB;
    const int e          = blockStart + t;

    // --- CDNA5 async global->LDS: stage this block's edge_attr_new slice ---
    // 256 edges * 3 floats = 768 floats = 192 x b128 transfers.
    {
        const int tile_floats = min(TPB * 3, (E_new - blockStart) * 3);
        if (t * 4 < tile_floats) {
            unsigned lds_addr = (unsigned)(uintptr_t)(&tile[0]) + (unsigned)(t * 16);
            unsigned g_off    = (unsigned)(t * 16);
            unsigned long long base =
                (unsigned long long)(uintptr_t)(attr_new + (size_t)blockStart * 3);
            asm volatile("global_load_async_to_lds_b128 %0, %1, %2"
                         :
                         : "v"(lds_addr), "v"(g_off), "s"(base)
                         : "memory");
        }
        asm volatile("s_wait_asynccnt 0" ::: "memory");
        __syncthreads();
    }

    if (e >= E_new) return;

    // --- random-access portion: hash probe + gather (latency hidden by
    //     many independent waves; one edge per lane) ---
    const int N   = *nptr;
    const int src = ei_new[e];
    const int dst = ei_new[E_new + e];
    const int key = src * N + dst;

    int match = -1;
    unsigned s = hash_mix(key) & hmask;
    for (unsigned probes = 0; probes <= hmask; ++probes) {
        const int k = hkey[s];
        if (k == key) { match = hval[s]; break; }
        if (k == -1)  break;
        s = (s + 1) & hmask;
    }

    float o0 = 0.f, o1 = 0.f, o2 = 0.f, o3 = 0.f, is_new = 1.f;
    if (match >= 0) {
        const float* r = attr_old + (size_t)match * 3;
        o0 = r[0];
        o1 = r[1];
        o2 = r[2];
        o3 = flow_old[match];
        is_new = 0.f;
    }

    const float a0 = tile[t * 3 + 0];
    const float a1 = tile[t * 3 + 1];
    const float a2 = tile[t * 3 + 2];

    // Two b128 stores per row: wave32 writes 1KB contiguous.
    float4* op = (float4*)(out + (size_t)e * 8);
    op[0] = make_float4(o0, o1, o2, o3);
    op[1] = make_float4(a0, a1, a2, is_new);
}

// ---------------------------------------------------------------------------
extern "C" void kernel_launch(void* const* d_in, const int* in_sizes, int n_in,
                              void* d_out, int out_size, void* d_ws, size_t ws_size,
                              hipStream_t stream) {
    const int*   ei_old   = (const int*)  d_in[0];  // [2, E_old]
    const float* attr_old = (const float*)d_in[1];  // [E_old, 3]
    const float* flow_old = (const float*)d_in[2];  // [E_old, 1]
    const int*   ei_new   = (const int*)  d_in[3];  // [2, E_new]
    const float* attr_new = (const float*)d_in[4];  // [E_new, 3]
    const int*   nptr     = (const int*)  d_in[5];  // scalar total_nodes
    float*       out      = (float*)d_out;

    const int E_old = in_sizes[0] / 2;
    const int E_new = in_sizes[3] / 2;

    // Hash table sizing: prefer 2^20 slots (8MB); shrink to fit ws but keep
    // at least 2^19 slots (floor of the shrink loop).
    unsigned hsize = 1u << 20;
    {
        const size_t max_slots = ws_size / (2 * sizeof(int));
        while (hsize > max_slots && hsize > (1u << 19)) hsize >>= 1;
    }
    const unsigned hmask = hsize - 1u;
    int* hkey = (int*)d_ws;
    int* hval = hkey + hsize;

    // 1) reset table
    {
        const int n4 = (int)(2u * hsize / 4u);
        hash_init_kernel<<<(n4 + TPB - 1) / TPB, TPB, 0, stream>>>((int4*)d_ws, n4);
    }
    // 2) insert old edges
    hash_insert_kernel<<<(E_old + TPB - 1) / TPB, TPB, 0, stream>>>(
        ei_old, nptr, E_old, hkey, hval, hmask);
    // 3) lookup + emit
    edge_align_kernel<<<(E_new + TPB - 1) / TPB, TPB, 0, stream>>>(
        ei_new, attr_new, attr_old, flow_old, hkey, hval, nptr, E_new, hmask, out);
}